// SentenceAttentionNetwork_8916352106827
// MI455X (gfx1250) — compile-verified
//
#include <hip/hip_runtime.h>
#include <hip/hip_bf16.h>
#include <stdint.h>

#define BB   2048
#define SS   512
#define EE   384
#define FF   28
#define HH   32
#define S1   513   // S+1 positions (prepended normed trainable row)

typedef __attribute__((ext_vector_type(2))) float v2f;
typedef __attribute__((ext_vector_type(8))) float v8f;

// Issue one 1536B embedding row as 3 async b128 copies (32 lanes x 16B each;
// INST_OFFSET applies to both the global and LDS address per CDNA5 ISA 10.x).
#define ASYNC_ROW(GA, LA)                                                      \
    asm volatile("global_load_async_to_lds_b128 %0, %1, off\n\t"               \
                 "global_load_async_to_lds_b128 %0, %1, off offset:512\n\t"    \
                 "global_load_async_to_lds_b128 %0, %1, off offset:1024"       \
                 :: "v"(LA), "v"(GA) : "memory")

// ---------------------------------------------------------------------------
// Kernel A: per-16-row prep, WMMA f32 16x16x4.
//   qmap[b,32] = features[b,:] @ W_map + b_map          (query / feature_map)
//   wq  [b,384] = W_k @ qmap[b,:]   (so scores = emb . wq, K-GEMM eliminated)
//   qbk [b]     = qmap[b,:] . b_k   (constant score bias, softmax-invariant)
//   nvec[384]   = trainable / ||trainable||   (block 0 only)
// One wave (32 threads) per 16 batch rows.
// ---------------------------------------------------------------------------
__global__ __launch_bounds__(32) void prep_kernel(
    const float* __restrict__ features, const float* __restrict__ W_map,
    const float* __restrict__ b_map,   const float* __restrict__ W_k,
    const float* __restrict__ b_k,     const float* __restrict__ trainable,
    float* __restrict__ qmap, float* __restrict__ wq,
    float* __restrict__ qbk,  float* __restrict__ nvec)
{
    const int lane  = threadIdx.x;        // 0..31
    const int base  = blockIdx.x * 16;    // batch row base
    const int m     = lane & 15;          // M index / N index within fragment
    const int khalf = lane >> 4;          // 0: K%4 in {0,1}, 1: K%4 in {2,3}

    __shared__ float Ft[16][FF];          // features tile
    __shared__ float Q[16][HH];           // feature_map tile (A of 2nd GEMM)

    for (int i = lane; i < 16 * FF; i += 32) {
        int r = i / FF, c = i % FF;
        Ft[r][c] = features[(size_t)(base + r) * FF + c];
    }
    __syncthreads();

    // ---- GEMM 1: [16 x 28] @ [28 x 32] via 16x16x4 f32 WMMA (7 K-steps) ----
    for (int nt = 0; nt < 2; ++nt) {                 // two N-tiles of 16 (H=32)
        v8f acc = {};
        for (int k = 0; k < 7; ++k) {                // K chunks of 4 (F=28)
            const int k0 = 4 * k + khalf * 2;
            v2f a, bf;
            a[0]  = Ft[m][k0 + 0];
            a[1]  = Ft[m][k0 + 1];
            bf[0] = W_map[(size_t)(k0 + 0) * HH + nt * 16 + m];
            bf[1] = W_map[(size_t)(k0 + 1) * HH + nt * 16 + m];
            acc = __builtin_amdgcn_wmma_f32_16x16x4_f32(
                false, a, false, bf, (short)0, acc, false, false);
        }
#pragma unroll
        for (int r = 0; r < 8; ++r) {                // D: vgpr r -> M=r (+8 hi half)
            const int M = r + khalf * 8;
            const float v = acc[r] + b_map[nt * 16 + m];
            Q[M][nt * 16 + m] = v;
            qmap[(size_t)(base + M) * HH + nt * 16 + m] = v;
        }
    }
    __syncthreads();

    // ---- GEMM 2: wq_tile[16 x 384] = Q[16 x 32] @ W_k^T[32 x 384] ----------
    // B[k=h][n=e] = W_k[e*H + h]
    for (int nt = 0; nt < 24; ++nt) {                // 24 N-tiles of 16 (E=384)
        v8f acc = {};
        const int e = nt * 16 + m;
        for (int k = 0; k < 8; ++k) {                // 8 K chunks of 4 (H=32)
            const int k0 = 4 * k + khalf * 2;
            v2f a, bf;
            a[0]  = Q[m][k0 + 0];
            a[1]  = Q[m][k0 + 1];
            bf[0] = W_k[(size_t)e * HH + k0 + 0];
            bf[1] = W_k[(size_t)e * HH + k0 + 1];
            acc = __builtin_amdgcn_wmma_f32_16x16x4_f32(
                false, a, false, bf, (short)0, acc, false, false);
        }
#pragma unroll
        for (int r = 0; r < 8; ++r) {
            const int M = r + khalf * 8;
            wq[(size_t)(base + M) * EE + nt * 16 + m] = acc[r];
        }
    }

    // ---- qbk[b] = Q[b,:] . b_k ---------------------------------------------
    if (lane < 16) {
        float s = 0.f;
        for (int h = 0; h < HH; ++h) s += Q[lane][h] * b_k[h];
        qbk[base + lane] = s;
    }

    // ---- nvec = trainable / ||trainable||  (block 0 only) ------------------
    if (blockIdx.x == 0) {
        float sq = 0.f;
        for (int j = lane; j < EE; j += 32) { float t = trainable[j]; sq += t * t; }
#pragma unroll
        for (int off = 16; off > 0; off >>= 1) sq += __shfl_xor(sq, off, 32);
        const float inv = rsqrtf(sq);
        for (int j = lane; j < EE; j += 32) nvec[j] = trainable[j] * inv;
    }
}

// ---------------------------------------------------------------------------
// Kernel B: one workgroup (256 threads = 8 waves) per batch row.
// Embedding rows are streamed HBM -> LDS with GLOBAL_LOAD_ASYNC_TO_LDS_B128
// (per-wave double buffer, ASYNCcnt-gated), then consumed from LDS for the
// online-softmax pass:
//   score_p = emb_p . wq_b * scale + qbk_b   (masked -> -inf)
//   acc[384] accumulates exp(score - m_run) * emb_p  (rescaled on new max)
// 8-wave merge, attn weights from stashed raw scores, doc_repr via 384x32
// matvec, logits via W_c. Memory-bound: embeddings read exactly once.
// ---------------------------------------------------------------------------
__global__ __launch_bounds__(256) void attn_kernel(
    const float* __restrict__ emb, const int* __restrict__ masks,
    const float* __restrict__ W_v, const float* __restrict__ b_v,
    const float* __restrict__ W_c, const float* __restrict__ b_c,
    const float* __restrict__ qmap, const float* __restrict__ wq,
    const float* __restrict__ qbk,  const float* __restrict__ nvec,
    float* __restrict__ out_logits, float* __restrict__ out_attn)
{
    const int b    = blockIdx.x;
    const int tid  = threadIdx.x;
    const int wave = tid >> 5;
    const int lane = tid & 31;

    __shared__ float wv[EE];        // wq row
    __shared__ float nv[EE];        // normed trainable (position 0 row)
    __shared__ int   msk[SS];       // mask row
    __shared__ float sc[S1];        // raw scores
    __shared__ float wm[8], wl[8];  // per-wave running max / sum
    __shared__ float wacc[8][EE];   // per-wave weighted-emb accumulators
    __shared__ float ev[EE];        // final normalized weighted embedding
    __shared__ __align__(16) float abuf[8 * 2 * EE];  // async stream buffers

    for (int j = tid; j < EE; j += 256) {
        wv[j] = wq[(size_t)b * EE + j];
        nv[j] = nvec[j];
    }
    for (int j = tid; j < SS; j += 256)
        msk[j] = masks[(size_t)b * SS + j];
    __syncthreads();

    const float scale = 0.17677669529663687f;   // 1/sqrt(32)
    const float qb    = qbk[b];
    const float* __restrict__ embB = emb + (size_t)b * SS * EE;
    const int c0 = lane * 12;                   // each lane owns 12 of 384 cols

    float m_run = -INFINITY, l_run = 0.f;
    float acc[12];
#pragma unroll
    for (int j = 0; j < 12; ++j) acc[j] = 0.f;

    // Every wave streams exactly 64 rows: positions p = start + 8k, k=0..63.
    // (wave 0 additionally handles the prepended position 0 from nv.)
    const int start = (wave == 0) ? 8 : wave;
    float* lbuf = abuf + wave * 2 * EE;         // two 1536B slots per wave
    const uint32_t lds0 = (uint32_t)(uintptr_t)lbuf + (uint32_t)(lane * 16);
    uint64_t ga = (uint64_t)(uintptr_t)embB
                + (uint64_t)(start - 1) * (EE * 4) + (uint64_t)(lane * 16);

    ASYNC_ROW(ga, lds0);                        // prime buffer 0

    if (wave == 0) {                            // position 0: normed trainable
        float er0[12], part = 0.f;
#pragma unroll
        for (int j = 0; j < 12; ++j) { er0[j] = nv[c0 + j]; part += er0[j] * wv[c0 + j]; }
#pragma unroll
        for (int off = 16; off > 0; off >>= 1) part += __shfl_xor(part, off, 32);
        const float s = part * scale + qb;
        if (lane == 0) sc[0] = s;
        m_run = s; l_run = 1.f;                 // exp(s - s) = 1
#pragma unroll
        for (int j = 0; j < 12; ++j) acc[j] = er0[j];
    }

    for (int k = 0; k < 64; ++k) {
        if (k + 1 < 64) {                       // issue next row, keep 3 in flight
            ga += 8ull * EE * 4;
            const uint32_t nxt = lds0 + (uint32_t)(((k + 1) & 1) * (EE * 4));
            ASYNC_ROW(ga, nxt);
            asm volatile("s_wait_asynccnt 0x3" ::: "memory");
        } else {
            asm volatile("s_wait_asynccnt 0x0" ::: "memory");
        }

        const float* __restrict__ row = lbuf + (k & 1) * EE;
        const float4 v0 = *(const float4*)(row + c0);
        const float4 v1 = *(const float4*)(row + c0 + 4);
        const float4 v2 = *(const float4*)(row + c0 + 8);
        float er[12];
        er[0] = v0.x; er[1] = v0.y; er[2]  = v0.z; er[3]  = v0.w;
        er[4] = v1.x; er[5] = v1.y; er[6]  = v1.z; er[7]  = v1.w;
        er[8] = v2.x; er[9] = v2.y; er[10] = v2.z; er[11] = v2.w;

        const int p = start + 8 * k;
        const bool masked = (msk[p - 1] != 0);

        float part = 0.f;
#pragma unroll
        for (int j = 0; j < 12; ++j) part += er[j] * wv[c0 + j];
#pragma unroll
        for (int off = 16; off > 0; off >>= 1) part += __shfl_xor(part, off, 32);

        const float s = masked ? -INFINITY : (part * scale + qb);
        if (lane == 0) sc[p] = s;

        if (!masked) {
            const float m_new = fmaxf(m_run, s);
            const float alpha = __expf(m_run - m_new);   // exp(-inf)=0 first time
            const float pe    = __expf(s - m_new);
#pragma unroll
            for (int j = 0; j < 12; ++j) acc[j] = acc[j] * alpha + pe * er[j];
            l_run = l_run * alpha + pe;
            m_run = m_new;
        }
    }

    if (lane == 0) { wm[wave] = m_run; wl[wave] = l_run; }
#pragma unroll
    for (int j = 0; j < 12; ++j) wacc[wave][c0 + j] = acc[j];
    __syncthreads();

    // ---- merge the 8 waves (wave 0 always saw unmasked p=0 => M finite) ----
    float M = -INFINITY;
#pragma unroll
    for (int w = 0; w < 8; ++w) M = fmaxf(M, wm[w]);
    float L = 0.f;
#pragma unroll
    for (int w = 0; w < 8; ++w) L += wl[w] * __expf(wm[w] - M);
    const float invL = 1.0f / L;

    for (int j = tid; j < EE; j += 256) {
        float t = 0.f;
#pragma unroll
        for (int w = 0; w < 8; ++w) t += wacc[w][j] * __expf(wm[w] - M);
        ev[j] = t * invL;
    }
    __syncthreads();

    // ---- attention weights out --------------------------------------------
    for (int p = tid; p < S1; p += 256)
        out_attn[(size_t)b * S1 + p] = __expf(sc[p] - M) * invL;   // -inf -> 0

    // ---- doc_repr = ev @ W_v + b_v ; logits = [doc_repr, q] @ W_c + b_c ----
    if (wave == 0) {
        float d = b_v[lane];
        for (int j = 0; j < EE; ++j) d += ev[j] * W_v[(size_t)j * HH + lane];
        const float q = qmap[(size_t)b * HH + lane];
        float c = d * W_c[lane] + q * W_c[HH + lane];
#pragma unroll
        for (int off = 16; off > 0; off >>= 1) c += __shfl_xor(c, off, 32);
        if (lane == 0) out_logits[b] = c + b_c[0];
    }
}

// ---------------------------------------------------------------------------
extern "C" void kernel_launch(void* const* d_in, const int* in_sizes, int n_in,
                              void* d_out, int out_size, void* d_ws, size_t ws_size,
                              hipStream_t stream)
{
    const float* emb       = (const float*)d_in[0];
    const int*   masks     = (const int*)  d_in[1];
    const float* features  = (const float*)d_in[2];
    const float* W_map     = (const float*)d_in[3];
    const float* b_map     = (const float*)d_in[4];
    const float* trainable = (const float*)d_in[5];
    const float* W_k       = (const float*)d_in[6];
    const float* b_k       = (const float*)d_in[7];
    const float* W_v       = (const float*)d_in[8];
    const float* b_v       = (const float*)d_in[9];
    const float* W_c       = (const float*)d_in[10];
    const float* b_c       = (const float*)d_in[11];

    float* out = (float*)d_out;            // [0,B): logits ; [B, B + B*S1): attn

    float* qmap = (float*)d_ws;            // B*H
    float* wqv  = qmap + (size_t)BB * HH;  // B*E
    float* qbk  = wqv  + (size_t)BB * EE;  // B
    float* nvec = qbk  + BB;               // E

    prep_kernel<<<BB / 16, 32, 0, stream>>>(features, W_map, b_map, W_k, b_k,
                                            trainable, qmap, wqv, qbk, nvec);
    attn_kernel<<<BB, 256, 0, stream>>>(emb, masks, W_v, b_v, W_c, b_c,
                                        qmap, wqv, qbk, nvec,
                                        out, out + BB);
}